// MultichannelMultiheadAttention_10299331576273
// MI455X (gfx1250) — compile-verified
//
#include <hip/hip_runtime.h>
#include <cstdint>

typedef float v2f __attribute__((ext_vector_type(2)));
typedef float v8f __attribute__((ext_vector_type(8)));

#define WMMA_F32X4(a, b, c) \
  __builtin_amdgcn_wmma_f32_16x16x4_f32(false, (a), false, (b), (short)0, (c), false, false)

constexpr int Bn = 4, Cc = 8, Hh = 256, Ww = 512, NH = 8, HD = 32;
constexpr size_t HW  = (size_t)Hh * Ww;        // 131072
constexpr size_t WH  = (size_t)Ww * HD;        // 16384 per (b,c,nh)
constexpr size_t QK_SLICE = (size_t)Ww * Ww;   // 262144
constexpr size_t OUT_ELEMS = (size_t)Bn * Cc * Hh * Ww;  // 4194304

// -ln(10000)/32 : rotary inv-freq exponent scale
#define ROT_SCALE (-0.28782313662425573f)

// ---------------------------------------------------------------------------
// Kernel 1: conv3x3 + positionwise 256x256 matmul (WMMA) + rotary, per
// (proj, b, c, 32-col tile). Q,V stored as (b,c,nh,w,hd); K stored as
// (b,c,h,w) so it is directly the B-operand (K^T) layout for QK^T.
// ---------------------------------------------------------------------------
__global__ void k_proj(const float* __restrict__ x,
                       const float* __restrict__ cwq, const float* __restrict__ cwk,
                       const float* __restrict__ cwv,
                       const float* __restrict__ pwq, const float* __restrict__ pwk,
                       const float* __restrict__ pwv,
                       float* __restrict__ Qt, float* __restrict__ Kz, float* __restrict__ Vt)
{
    extern __shared__ float smem[];
    float* ylds = smem;                 // [256][33] conv output tile
    float* xlds = smem + 256 * 33;      // [258][34] x input tile (one ic at a time)

    const int idx  = blockIdx.x;
    const int wt   = idx & 15;
    const int c    = (idx >> 4) & 7;
    const int b    = (idx >> 7) & 3;
    const int proj = idx >> 9;          // 0=q 1=k 2=v
    const int w0   = wt * 32;

    const float* cw = (proj == 0 ? cwq : (proj == 1 ? cwk : cwv)) + c * Cc * 9;
    const float* pw = (proj == 0 ? pwq : (proj == 1 ? pwk : pwv)) + (size_t)c * Hh * Hh;
    const float* xb = x + (size_t)b * Cc * HW;

    const int tid = threadIdx.x;
    const int h   = tid;                // each thread owns conv row h

    float acc[32];
#pragma unroll
    for (int j = 0; j < 32; ++j) acc[j] = 0.f;

    for (int ic = 0; ic < Cc; ++ic) {
        // stage x[b][ic] rows h=-1..256, cols w0-1..w0+32 (zero padded)
        for (int i = tid; i < 258 * 34; i += 256) {
            int r  = i / 34, cc2 = i - r * 34;
            int hh = r - 1,  ww  = w0 - 1 + cc2;
            float v = 0.f;
            if (hh >= 0 && hh < Hh && ww >= 0 && ww < Ww)
                v = xb[(size_t)ic * HW + (size_t)hh * Ww + ww];
            xlds[i] = v;
        }
        __syncthreads();
        float k9[9];
#pragma unroll
        for (int t = 0; t < 9; ++t) k9[t] = cw[ic * 9 + t];
#pragma unroll
        for (int dh = 0; dh < 3; ++dh) {
            const float* xr = xlds + (h + dh) * 34;
            float x0 = xr[0], x1 = xr[1];
#pragma unroll
            for (int j = 0; j < 32; ++j) {
                float x2 = xr[j + 2];
                acc[j] += k9[dh * 3 + 0] * x0 + k9[dh * 3 + 1] * x1 + k9[dh * 3 + 2] * x2;
                x0 = x1; x1 = x2;
            }
        }
        __syncthreads();
    }
#pragma unroll
    for (int j = 0; j < 32; ++j) ylds[h * 33 + j] = acc[j];
    __syncthreads();

    // ---- positionwise matmul: z[f,w] = sum_h pw[f,h] * y[h,w] via WMMA ----
    const int wid  = tid >> 5;
    const int lane = tid & 31;
    const int ln   = lane & 15;
    const int half = lane >> 4;

    for (int t = wid; t < 32; t += 8) {          // 16 f-tiles x 2 w-tiles
        const int f0  = (t >> 1) * 16;
        const int wc0 = (t & 1) * 16;
        v8f d = {0.f, 0.f, 0.f, 0.f, 0.f, 0.f, 0.f, 0.f};
#pragma unroll 8
        for (int kb = 0; kb < 256; kb += 4) {
            const int k = kb + 2 * half;
            v2f a = *(const v2f*)(pw + (size_t)(f0 + ln) * Hh + k);
            v2f bv;
            bv.x = ylds[k * 33 + wc0 + ln];
            bv.y = ylds[(k + 1) * 33 + wc0 + ln];
            d = WMMA_F32X4(a, bv, d);
        }
        const int fbase = f0 + half * 8;         // 8 consecutive f rows in this lane
        const int wcol  = w0 + wc0 + ln;         // sequence position
        if (proj < 2) {                          // rotary on q,k (interleaved pairs)
#pragma unroll
            for (int p = 0; p < 4; ++p) {
                int dim = (fbase + 2 * p) & 31;
                // 10000^(-dim/32) = exp(dim * -ln(1e4)/32): one transcendental
                float th = (float)wcol * __expf((float)dim * ROT_SCALE);
                float cs = __cosf(th), sn = __sinf(th);
                float e = d[2 * p], o = d[2 * p + 1];
                d[2 * p]     = e * cs - o * sn;
                d[2 * p + 1] = o * cs + e * sn;
            }
        }
        if (proj != 1) {                         // Q or V -> (b,c,nh,w,hd)
            const int nh = fbase >> 5, db = fbase & 31;
            float* dst = (proj == 0 ? Qt : Vt) +
                         (((size_t)(b * Cc + c) * NH + nh) * Ww + wcol) * HD + db;
            *(float4*)(dst)     = make_float4(d[0], d[1], d[2], d[3]);
            *(float4*)(dst + 4) = make_float4(d[4], d[5], d[6], d[7]);
        } else {                                 // K -> (b,c,h,w)
            size_t base = ((size_t)(b * Cc + c) * Hh + fbase) * Ww + wcol;
#pragma unroll
            for (int j = 0; j < 8; ++j) Kz[base + (size_t)j * Ww] = d[j];
        }
    }
}

// ---------------------------------------------------------------------------
// Kernel 2: per (b,c,nh, 64-row q block): S = QK^T/16 + pqk (write qk),
// softmax in LDS, O = P*V. 8 waves, 64x516 f32 LDS score buffer (132KB).
// pqk/qk are streamed once (536MB total) -> non-temporal so the reused
// Q/K/V/A intermediates (64MB) stay resident in the 192MB L2.
// ---------------------------------------------------------------------------
__global__ void k_attn(const float* __restrict__ Qt, const float* __restrict__ Kz,
                       const float* __restrict__ Vt, const float* __restrict__ pqk,
                       float* __restrict__ qk, float* __restrict__ At)
{
    extern __shared__ float S[];                 // [64][516]
    const int idx = blockIdx.x;
    const int qb = idx & 7;
    const int nh = (idx >> 3) & 7;
    const int c  = (idx >> 6) & 7;
    const int b  = idx >> 9;
    const int q0 = qb * 64;

    const size_t bcn = (size_t)(b * Cc + c) * NH + nh;
    const float* Qp = Qt + bcn * WH;
    const float* Kp = Kz + ((size_t)(b * Cc + c) * Hh + nh * HD) * Ww;
    const float* Vp = Vt + bcn * WH;
    const float* pq = pqk + bcn * QK_SLICE;
    float* qko = qk + bcn * QK_SLICE;

    const int tid = threadIdx.x, wid = tid >> 5, lane = tid & 31;
    const int ln = lane & 15, half = lane >> 4;

    // ---- phase 1: S = QK^T * (1/16) + pqk ----
    for (int t = wid; t < 128; t += 8) {         // 4 q-tiles x 32 k-tiles
        const int qt = t >> 5, kt = t & 31;
        const int rowb = q0 + qt * 16, col0 = kt * 16;
        v8f d = {0.f, 0.f, 0.f, 0.f, 0.f, 0.f, 0.f, 0.f};
#pragma unroll
        for (int kb = 0; kb < 32; kb += 4) {     // hd = 32
            const int k = kb + 2 * half;
            v2f a = *(const v2f*)(Qp + (size_t)(rowb + ln) * HD + k);
            v2f bv;
            bv.x = Kp[(size_t)k * Ww + col0 + ln];
            bv.y = Kp[(size_t)(k + 1) * Ww + col0 + ln];
            d = WMMA_F32X4(a, bv, d);
        }
#pragma unroll
        for (int j = 0; j < 8; ++j) {
            const int row = rowb + j + 8 * half;
            const int col = col0 + ln;
            // stream-once traffic: non-temporal load & store (TH=NT)
            float v = d[j] * 0.0625f +
                      __builtin_nontemporal_load(&pq[(size_t)row * Ww + col]);
            __builtin_nontemporal_store(v, &qko[(size_t)row * Ww + col]);
            S[(qt * 16 + j + 8 * half) * 516 + col] = v;
        }
    }
    __syncthreads();

    // ---- phase 2: row softmax (8 rows per wave) ----
    for (int r = 0; r < 8; ++r) {
        const int row = wid * 8 + r;
        float* Sr = S + row * 516;
        float mx = -3.4e38f;
        for (int i = lane; i < Ww; i += 32) mx = fmaxf(mx, Sr[i]);
#pragma unroll
        for (int off = 16; off > 0; off >>= 1) mx = fmaxf(mx, __shfl_xor(mx, off, 32));
        float sum = 0.f;
        for (int i = lane; i < Ww; i += 32) { float e = __expf(Sr[i] - mx); Sr[i] = e; sum += e; }
#pragma unroll
        for (int off = 16; off > 0; off >>= 1) sum += __shfl_xor(sum, off, 32);
        const float inv = 1.0f / sum;
        for (int i = lane; i < Ww; i += 32) Sr[i] *= inv;
    }
    __syncthreads();

    // ---- phase 3: O = P @ V (64x32, one 16x16 tile per wave) ----
    const int qt2 = wid >> 1, dt = wid & 1;
    v8f d = {0.f, 0.f, 0.f, 0.f, 0.f, 0.f, 0.f, 0.f};
#pragma unroll 4
    for (int kb = 0; kb < Ww; kb += 4) {
        const int k = kb + 2 * half;
        v2f a = *(const v2f*)(S + (qt2 * 16 + ln) * 516 + k);
        v2f bv;
        bv.x = Vp[(size_t)k * HD + dt * 16 + ln];
        bv.y = Vp[(size_t)(k + 1) * HD + dt * 16 + ln];
        d = WMMA_F32X4(a, bv, d);
    }
    float* Ao = At + bcn * WH;
#pragma unroll
    for (int j = 0; j < 8; ++j) {
        const int row = q0 + qt2 * 16 + j + 8 * half;
        Ao[(size_t)row * HD + dt * 16 + ln] = d[j];
    }
}

// ---------------------------------------------------------------------------
// Kernel 3: out1[f,w] = sum_h pw_out[f,h] * a_z[h,w] via WMMA, then
// out[o] = sum_c dw_out[o,c] * out1[c], accumulated in registers.
// ---------------------------------------------------------------------------
__global__ void k_out(const float* __restrict__ At, const float* __restrict__ pwo,
                      const float* __restrict__ dwo, float* __restrict__ out)
{
    const int tid = threadIdx.x, wid = tid >> 5, lane = tid & 31;
    const int ln = lane & 15, half = lane >> 4;
    const int g  = blockIdx.x * 8 + wid;         // 2048 tiles
    const int wt = g & 31, ft = (g >> 5) & 15, b = g >> 9;
    const int w0 = wt * 16, f0 = ft * 16;

    v8f accv[8];
#pragma unroll
    for (int o = 0; o < 8; ++o) accv[o] = (v8f){0.f, 0.f, 0.f, 0.f, 0.f, 0.f, 0.f, 0.f};

    for (int c = 0; c < Cc; ++c) {
        const float* pwc = pwo + (size_t)c * Hh * Hh;
        const float* Ab  = At + (size_t)(b * Cc + c) * NH * WH;
        v8f d = {0.f, 0.f, 0.f, 0.f, 0.f, 0.f, 0.f, 0.f};
#pragma unroll 8
        for (int kb = 0; kb < 256; kb += 4) {
            const int k = kb + 2 * half;          // h index (even)
            v2f a = *(const v2f*)(pwc + (size_t)(f0 + ln) * Hh + k);
            const int nh0 = k >> 5, d0 = k & 31;  // k,k+1 stay in same head block
            v2f bv = *(const v2f*)(Ab + ((size_t)nh0 * Ww + (w0 + ln)) * HD + d0);
            d = WMMA_F32X4(a, bv, d);
        }
#pragma unroll
        for (int o = 0; o < 8; ++o) {
            float s = dwo[o * Cc + c];
            accv[o] += s * d;
        }
    }
#pragma unroll
    for (int o = 0; o < 8; ++o)
#pragma unroll
        for (int j = 0; j < 8; ++j) {
            const int f = f0 + j + 8 * half;
            out[((size_t)(b * Cc + o) * Hh + f) * Ww + w0 + ln] = accv[o][j];
        }
}

// ---------------------------------------------------------------------------
extern "C" void kernel_launch(void* const* d_in, const int* in_sizes, int n_in,
                              void* d_out, int out_size, void* d_ws, size_t ws_size,
                              hipStream_t stream)
{
    (void)in_sizes; (void)n_in; (void)out_size; (void)ws_size;
    const float* x   = (const float*)d_in[0];
    const float* pqk = (const float*)d_in[1];
    const float* cwq = (const float*)d_in[2];
    const float* cwk = (const float*)d_in[3];
    const float* cwv = (const float*)d_in[4];
    const float* pwq = (const float*)d_in[5];
    const float* pwk = (const float*)d_in[6];
    const float* pwv = (const float*)d_in[7];
    const float* pwo = (const float*)d_in[8];
    const float* dwo = (const float*)d_in[9];

    float* out = (float*)d_out;
    float* qk  = out + OUT_ELEMS;                // tuple (out, qk) concatenated

    float* ws = (float*)d_ws;                    // 4 x 16MB scratch
    float* Qt = ws;
    float* Kz = ws + OUT_ELEMS;
    float* Vt = ws + 2 * OUT_ELEMS;
    float* At = ws + 3 * OUT_ELEMS;

    // 3 proj x 4 b x 8 c x 16 w-tiles = 1536 WGs; LDS = y[256*33] + x[258*34]
    k_proj<<<dim3(1536), dim3(256), (256 * 33 + 258 * 34) * sizeof(float), stream>>>(
        x, cwq, cwk, cwv, pwq, pwk, pwv, Qt, Kz, Vt);
    // 4 b x 8 c x 8 nh x 8 q-blocks = 2048 WGs; LDS = 64 x 516 floats
    k_attn<<<dim3(2048), dim3(256), 64 * 516 * sizeof(float), stream>>>(
        Qt, Kz, Vt, pqk, qk, At);
    // 2048 output tiles / 8 waves = 256 WGs
    k_out<<<dim3(256), dim3(256), 0, stream>>>(At, pwo, dwo, out);
}